// UFLN_31988916420870
// MI455X (gfx1250) — compile-verified
//
#include <hip/hip_runtime.h>

// GCN-style two-branch network for MI455X (gfx1250, wave32).
// GEMMs use V_WMMA_F32_16X16X4_F32 (fp32 fidelity; GEMMs are ~19 GFLOP total,
// far below the wmma ceiling -> not the bottleneck). SpMM is edge-parallel
// with global_atomic_add_f32; gathered dense matrices (<=54 MB) are L2-resident
// (192 MB L2), so irregular traffic mostly stays on-chip.
//
// Workspace requirement: ~138 MB (see layout below).

#define N_NODES 50000
#define N_EDGES 800000

typedef float v2f __attribute__((ext_vector_type(2)));
typedef float v8f __attribute__((ext_vector_type(8)));

// ---------------------------------------------------------------------------
// Pack W1|W2|W3 -> [128 x 208], W4|W5 -> [204 x 272], Wm -> [128 x 144],
// zero padding in extra columns so padded GEMM outputs are zeros.
// ---------------------------------------------------------------------------
__global__ __launch_bounds__(256) void pack_weights_kernel(
    const float* __restrict__ W1, const float* __restrict__ W2,
    const float* __restrict__ W3, const float* __restrict__ W4,
    const float* __restrict__ W5, const float* __restrict__ Wm,
    float* __restrict__ wcat1, float* __restrict__ wcat45,
    float* __restrict__ wmp) {
    const int n1 = 128 * 208, n2 = 204 * 272, n3 = 128 * 144;
    int t = blockIdx.x * 256 + threadIdx.x;
    if (t < n1) {
        int k = t / 208, c = t % 208;
        float v = (c < 64)  ? W1[k * 64 + c]
                : (c < 132) ? W2[k * 68 + (c - 64)]
                : (c < 204) ? W3[k * 72 + (c - 132)] : 0.0f;
        wcat1[t] = v;
    } else if (t < n1 + n2) {
        int u = t - n1;
        int k = u / 272, c = u % 272;
        float v = (c < 132) ? W4[k * 132 + c]
                : (c < 260) ? W5[k * 128 + (c - 132)] : 0.0f;
        wcat45[u] = v;
    } else if (t < n1 + n2 + n3) {
        int u = t - n1 - n2;
        int k = u / 144, c = u % 144;
        wmp[u] = (c < 132) ? Wm[k * 132 + c] : 0.0f;
    }
}

// ---------------------------------------------------------------------------
// fp32 WMMA GEMM: C[M x Fpad] = A[M x K] @ B[K x Fpad].
// One wave per 16x16 C tile; K stepped by 4 via V_WMMA_F32_16X16X4_F32.
// A 16x4 layout: lanes 0-15 hold (M=lane, K=k..k+1), lanes 16-31 hold K=k+2..k+3.
// B 4x16 layout: lanes 0-15 hold (K=k..k+1, N=lane), lanes 16-31 K=k+2..k+3.
// C/D layout: VGPR i: lanes 0-15 -> M=i, lanes 16-31 -> M=8+i; N = lane&15.
// Tile guard is wave-uniform -> EXEC all ones inside WMMA (ISA requirement).
// ---------------------------------------------------------------------------
__global__ __launch_bounds__(256) void wmma_gemm_f32_kernel(
    const float* __restrict__ A, const float* __restrict__ B,
    float* __restrict__ C, int M, int K, int lda, int ldb, int ldc,
    int colTiles) {
    int gwave = (blockIdx.x * 256 + threadIdx.x) >> 5;
    int lane  = threadIdx.x & 31;
    int rowTiles = M >> 4;
    int rt = gwave / colTiles;
    int ct = gwave - rt * colTiles;
    if (rt >= rowTiles) return;                    // wave-uniform exit

    int mrow = (rt << 4) + (lane & 15);
    int ncol = (ct << 4) + (lane & 15);
    int koff = (lane >> 4) << 1;                   // 0 (lanes 0-15) or 2

    const float* arow = A + (long long)mrow * lda + koff;
    const float* bcol = B + (long long)koff * ldb + ncol;

    v8f acc = {};
    for (int k = 0; k < K; k += 4) {
        float2 av = *reinterpret_cast<const float2*>(arow + k); // 8B aligned (lda even, k+koff even)
        v2f a; a.x = av.x; a.y = av.y;
        v2f b;
        b.x = bcol[(long long)k * ldb];
        b.y = bcol[(long long)(k + 1) * ldb];
        acc = __builtin_amdgcn_wmma_f32_16x16x4_f32(
            false, a, false, b, (short)0, acc, false, false);
    }

    int mbase = (rt << 4) + ((lane >> 4) << 3);    // +0 or +8
    float* crow = C + (long long)mbase * ldc + ncol;
#pragma unroll
    for (int i = 0; i < 8; ++i) crow[(long long)i * ldc] = acc[i];
}

// ---------------------------------------------------------------------------
// Edge-parallel SpMM scatter: Out[row[e], :F] += val[e] * D[col[e], :F].
// One thread handles 4 features of one edge (float4 gather + 4 f32 atomics).
// Accumulator rows live in L2 (42/54 MB buffers) -> L2-resident atomics.
// ---------------------------------------------------------------------------
__global__ __launch_bounds__(256) void spmm_atomic_kernel(
    const int* __restrict__ idx,   // [2*E]: rows then cols
    const float* __restrict__ val, // [E]
    const float* __restrict__ D, float* __restrict__ Out,
    int E, int chunks, int ldd, int ldo) {
    long long t = (long long)blockIdx.x * 256 + threadIdx.x;
    long long total = (long long)E * chunks;
    if (t >= total) return;
    int e = (int)(t / chunks);
    int c = (int)(t - (long long)e * chunks) * 4;
    int r   = idx[e];
    int col = idx[E + e];
    float v = val[e];
    const float4 s = *reinterpret_cast<const float4*>(D + (long long)col * ldd + c);
    float* dst = Out + (long long)r * ldo + c;
    atomicAdd(dst + 0, v * s.x);
    atomicAdd(dst + 1, v * s.y);
    atomicAdd(dst + 2, v * s.z);
    atomicAdd(dst + 3, v * s.w);
}

// ---------------------------------------------------------------------------
// Stage-1 activation: one wave per node.
//   t[f]   = sigmoid(S1[i,f] + bcat(f)),  f < 204   (b1|b2|b3)
//   msec   = mean(t[64..131])             (68 cols of h_sec)
//   low[i,f] = f < 132 ? t[f] : msec * t[f]   -> d_out x_low/y_low [N,204]
// ---------------------------------------------------------------------------
__global__ __launch_bounds__(256) void act1_kernel(
    const float* __restrict__ S1, const float* __restrict__ b1,
    const float* __restrict__ b2, const float* __restrict__ b3,
    float* __restrict__ low) {
    int node = (blockIdx.x * 256 + threadIdx.x) >> 5;
    int lane = threadIdx.x & 31;
    if (node >= N_NODES) return;
    const float* srow = S1 + (long long)node * 208;
    float t[7];
    float sum_sec = 0.0f;
#pragma unroll
    for (int j = 0; j < 7; ++j) {
        int f = lane + 32 * j;
        float v = 0.0f;
        if (f < 204) {
            float bias = (f < 64) ? b1[f] : (f < 132) ? b2[f - 64] : b3[f - 132];
            v = 1.0f / (1.0f + expf(-(srow[f] + bias)));
            if (f >= 64 && f < 132) sum_sec += v;
        }
        t[j] = v;
    }
#pragma unroll
    for (int off = 16; off > 0; off >>= 1) sum_sec += __shfl_xor(sum_sec, off, 32);
    float msec = sum_sec * (1.0f / 68.0f);
    float* drow = low + (long long)node * 204;
#pragma unroll
    for (int j = 0; j < 7; ++j) {
        int f = lane + 32 * j;
        if (f < 204) drow[f] = (f < 132) ? t[j] : msec * t[j];
    }
}

// ---------------------------------------------------------------------------
// h_fiv = S2[:,132:260] + b5  -> d_out x_fiv/y_fiv [N,128]
// ---------------------------------------------------------------------------
__global__ __launch_bounds__(256) void bias_fiv_kernel(
    const float* __restrict__ S2, const float* __restrict__ b5,
    float* __restrict__ fiv) {
    int t = blockIdx.x * 256 + threadIdx.x;
    if (t >= N_NODES * 128) return;
    int i = t >> 7, f = t & 127;
    fiv[t] = S2[(long long)i * 272 + 132 + f] + b5[f];
}

// ---------------------------------------------------------------------------
// Finisher: one wave per node.
//   lr      = low_result row (204, already in d_out)
//   low     = mean(lr)*lr + lr                 -> final[0:204]
//   h_mlp   = leaky_relu(Mws[i,f]+bm[f],0.01)  -> d_out x_mlp/y_mlp [N,132]
//   h_fou   = S2[i,f]+b4[f]
//   h_f3    = (h_mlp+h_fou)/2                  -> final[204:336]
// ---------------------------------------------------------------------------
__global__ __launch_bounds__(256) void final_kernel(
    const float* __restrict__ low, const float* __restrict__ S2,
    const float* __restrict__ Mws, const float* __restrict__ b4,
    const float* __restrict__ bm, float* __restrict__ outFinal,
    float* __restrict__ outMlp) {
    int node = (blockIdx.x * 256 + threadIdx.x) >> 5;
    int lane = threadIdx.x & 31;
    if (node >= N_NODES) return;
    const float* lrow = low + (long long)node * 204;
    float lr[7];
    float s = 0.0f;
#pragma unroll
    for (int j = 0; j < 7; ++j) {
        int f = lane + 32 * j;
        lr[j] = (f < 204) ? lrow[f] : 0.0f;
        s += lr[j];
    }
#pragma unroll
    for (int off = 16; off > 0; off >>= 1) s += __shfl_xor(s, off, 32);
    float mean = s * (1.0f / 204.0f);
    float* frow = outFinal + (long long)node * 336;
#pragma unroll
    for (int j = 0; j < 7; ++j) {
        int f = lane + 32 * j;
        if (f < 204) frow[f] = mean * lr[j] + lr[j];
    }
    const float* s2row = S2 + (long long)node * 272;
    const float* mrow  = Mws + (long long)node * 144;
    float* mlprow = outMlp + (long long)node * 132;
#pragma unroll
    for (int j = 0; j < 5; ++j) {
        int f = lane + 32 * j;
        if (f < 132) {
            float mlp = mrow[f] + bm[f];
            mlp = (mlp > 0.0f) ? mlp : 0.01f * mlp;
            float fou = s2row[f] + b4[f];
            mlprow[f] = mlp;
            frow[204 + f] = 0.5f * (mlp + fou);
        }
    }
}

// ---------------------------------------------------------------------------
// Host side
// ---------------------------------------------------------------------------
static inline int ceil_div(long long a, long long b) { return (int)((a + b - 1) / b); }

static void run_branch(const float* h, const int* idx, const float* eval_,
                       const float* b1, const float* b2, const float* b3,
                       const float* b4, const float* b5, const float* bm,
                       const float* wcat1, const float* wcat45, const float* wmp,
                       float* bufA, float* bufB, float* bufM,
                       float* out_low, float* out_final, float* out_fiv,
                       float* out_mlp, hipStream_t stream) {
    const int N = N_NODES, E = N_EDGES;
    const int rowTiles = N / 16;  // 3125 exact

    // G1: P = h @ Wcat1  [N x 208]
    {
        int colTiles = 208 / 16;
        int waves = rowTiles * colTiles;
        wmma_gemm_f32_kernel<<<ceil_div(waves, 8), 256, 0, stream>>>(
            h, wcat1, bufA, N, 128, 128, 208, 208, colTiles);
    }
    // S1 = spmm(adj, P[:, :204])
    hipMemsetAsync(bufB, 0, (size_t)N * 208 * sizeof(float), stream);
    {
        long long total = (long long)E * 51;  // 204/4 chunks
        spmm_atomic_kernel<<<ceil_div(total, 256), 256, 0, stream>>>(
            idx, eval_, bufA, bufB, E, 51, 208, 208);
    }
    // low_result -> out_low [N,204]
    act1_kernel<<<ceil_div((long long)N * 32, 256), 256, 0, stream>>>(
        bufB, b1, b2, b3, out_low);

    // G2: Q = low @ Wcat45  [N x 272]
    {
        int colTiles = 272 / 16;
        int waves = rowTiles * colTiles;
        wmma_gemm_f32_kernel<<<ceil_div(waves, 8), 256, 0, stream>>>(
            out_low, wcat45, bufA, N, 204, 204, 272, 272, colTiles);
    }
    // S2 = spmm(adj, Q[:, :260])
    hipMemsetAsync(bufB, 0, (size_t)N * 272 * sizeof(float), stream);
    {
        long long total = (long long)E * 65;  // 260/4 chunks
        spmm_atomic_kernel<<<ceil_div(total, 256), 256, 0, stream>>>(
            idx, eval_, bufA, bufB, E, 65, 272, 272);
    }
    // h_fiv -> out_fiv [N,128]
    bias_fiv_kernel<<<ceil_div((long long)N * 128, 256), 256, 0, stream>>>(
        bufB, b5, out_fiv);

    // G3: Mws = h_fiv @ Wm_p  [N x 144]
    {
        int colTiles = 144 / 16;
        int waves = rowTiles * colTiles;
        wmma_gemm_f32_kernel<<<ceil_div(waves, 8), 256, 0, stream>>>(
            out_fiv, wmp, bufM, N, 128, 128, 144, 144, colTiles);
    }
    // finisher -> out_final [N,336], out_mlp [N,132]
    final_kernel<<<ceil_div((long long)N * 32, 256), 256, 0, stream>>>(
        out_low, bufB, bufM, b4, bm, out_final, out_mlp);
}

extern "C" void kernel_launch(void* const* d_in, const int* in_sizes, int n_in,
                              void* d_out, int out_size, void* d_ws, size_t ws_size,
                              hipStream_t stream) {
    const float* x       = (const float*)d_in[0];
    const int*   adj1    = (const int*)d_in[1];
    const float* a1val   = (const float*)d_in[2];
    const float* y       = (const float*)d_in[3];
    const int*   adj2    = (const int*)d_in[4];
    const float* a2val   = (const float*)d_in[5];
    const float* W1 = (const float*)d_in[6];
    const float* b1 = (const float*)d_in[7];
    const float* W2 = (const float*)d_in[8];
    const float* b2 = (const float*)d_in[9];
    const float* W3 = (const float*)d_in[10];
    const float* b3 = (const float*)d_in[11];
    const float* W4 = (const float*)d_in[12];
    const float* b4 = (const float*)d_in[13];
    const float* W5 = (const float*)d_in[14];
    const float* b5 = (const float*)d_in[15];
    const float* Wm = (const float*)d_in[16];
    const float* bm = (const float*)d_in[17];

    const long long N = N_NODES;
    float* out = (float*)d_out;
    // output regions (return order)
    float* x_low   = out;
    float* y_low   = out + N * 204;
    float* x_final = out + 2 * N * 204;
    float* y_final = out + 2 * N * 204 + N * 336;
    float* x_fiv   = out + 2 * N * 204 + 2 * N * 336;
    float* x_mlp   = x_fiv + N * 128;
    float* y_fiv   = x_mlp + N * 132;
    float* y_mlp   = y_fiv + N * 128;

    // workspace layout (floats); requires ~138 MB
    float* ws = (float*)d_ws;
    float* wcat1  = ws;                       // 128*208 = 26624
    float* wcat45 = wcat1 + 26624;            // 204*272 = 55488
    float* wmp    = wcat45 + 55488;           // 128*144 = 18432
    float* bufA   = wmp + 18432;              // max(N*208, N*272) = 13,600,000 (P/Q)
    float* bufB   = bufA + 13600000;          // 13,600,000 (S1/S2 atomic accum)
    float* bufM   = bufB + 13600000;          // N*144 = 7,200,000 (MLP pre-act)

    // pack weights once per launch (deterministic)
    {
        int total = 128 * 208 + 204 * 272 + 128 * 144;
        pack_weights_kernel<<<ceil_div(total, 256), 256, 0, stream>>>(
            W1, W2, W3, W4, W5, Wm, wcat1, wcat45, wmp);
    }

    run_branch(x, adj1, a1val, b1, b2, b3, b4, b5, bm,
               wcat1, wcat45, wmp, bufA, bufB, bufM,
               x_low, x_final, x_fiv, x_mlp, stream);
    run_branch(y, adj2, a2val, b1, b2, b3, b4, b5, bm,
               wcat1, wcat45, wmp, bufA, bufB, bufM,
               y_low, y_final, y_fiv, y_mlp, stream);
}